// GraphAttentionLayer_13924283973819
// MI455X (gfx1250) — compile-verified
//
#include <hip/hip_runtime.h>
#include <hip/hip_bf16.h>
#include <math.h>

typedef __attribute__((ext_vector_type(2))) float v2f;
typedef __attribute__((ext_vector_type(8))) float v8f;

#define B_   4
#define N_   2048
#define FIN  256
#define FOUT 128
#define ALPHA_ 0.2f
#define EPS_   1e-5f
#define NEG_   -9000000000000000.0f

// ---------------- Kernel 1: Wh = h @ W  via V_WMMA_F32_16X16X4_F32 ----------------
// One wave per 16x16 output tile. 4096 tiles total, 4 waves/block.
__global__ __launch_bounds__(128) void wh_kernel(const float* __restrict__ h,
                                                 const float* __restrict__ W,
                                                 float* __restrict__ Wh) {
  int lane  = threadIdx.x & 31;
  int tile  = blockIdx.x * 4 + (threadIdx.x >> 5);
  int otile = tile & 7;
  int rest  = tile >> 3;
  int itile = rest & (N_ / 16 - 1);
  int b     = rest >> 7;
  int i0 = itile * 16, o0 = otile * 16;
  int r16  = lane & 15;            // M for A-frag, N for B-frag
  int koff = (lane >> 4) << 1;     // 0 or 2

  const float* hrow = h + (size_t)(b * N_ + i0 + r16) * FIN;
  v8f c = {};
  for (int k0 = 0; k0 < FIN; k0 += 4) {
    float2 av = *(const float2*)(hrow + k0 + koff);
    v2f a; a.x = av.x; a.y = av.y;
    v2f bb;
    bb.x = W[(size_t)(k0 + koff + 0) * FOUT + o0 + r16];
    bb.y = W[(size_t)(k0 + koff + 1) * FOUT + o0 + r16];
    c = __builtin_amdgcn_wmma_f32_16x16x4_f32(false, a, false, bb, (short)0, c,
                                              false, false);
  }
#pragma unroll
  for (int r = 0; r < 8; ++r) {
    int M = r + ((lane >> 4) << 3);
    Wh[(size_t)(b * N_ + i0 + M) * FOUT + o0 + r16] = c[r];
  }
}

// ---------------- Kernel 2: s1 = Wh@a1, s2 = Wh@a2 (wave per node) ----------------
__global__ __launch_bounds__(256) void s12_kernel(const float* __restrict__ Wh,
                                                  const float* __restrict__ a,
                                                  float* __restrict__ s1,
                                                  float* __restrict__ s2) {
  int lane = threadIdx.x & 31;
  int node = (int)((blockIdx.x * blockDim.x + threadIdx.x) >> 5);
  if (node >= B_ * N_) return;
  const float4* w4 = (const float4*)(Wh + (size_t)node * FOUT);
  const float4* a1 = (const float4*)a;
  const float4* a2 = (const float4*)(a + FOUT);
  float4 wv = w4[lane];
  float4 x1 = a1[lane];
  float4 x2 = a2[lane];
  float p1 = wv.x * x1.x + wv.y * x1.y + wv.z * x1.z + wv.w * x1.w;
  float p2 = wv.x * x2.x + wv.y * x2.y + wv.z * x2.z + wv.w * x2.w;
  for (int off = 16; off >= 1; off >>= 1) {
    p1 += __shfl_xor(p1, off, 32);
    p2 += __shfl_xor(p2, off, 32);
  }
  if (lane == 0) { s1[node] = p1; s2[node] = p2; }
}

// -------- Kernel 3: online softmax stats per row (wave per row) + pack adj --------
// Single pass over the 67MB adj tensor; emits a 2MB bitmask for the WMMA pass.
__global__ __launch_bounds__(256) void stats_kernel(const int* __restrict__ adj,
                                                    const float* __restrict__ s1,
                                                    const float* __restrict__ s2,
                                                    float* __restrict__ mrow,
                                                    float* __restrict__ lrow,
                                                    unsigned* __restrict__ mbits) {
  int lane = threadIdx.x & 31;
  int row  = blockIdx.x * 8 + (threadIdx.x >> 5);   // b*N + i
  int b    = row >> 11;                             // / N_
  float s1v = s1[row];
  const float* s2b = s2 + (size_t)b * N_;
  const int* arow  = adj + (size_t)row * N_;
  unsigned* mb     = mbits + (size_t)row * (N_ / 32);
  float m = -INFINITY, l = 0.f;
  for (int w = 0; w < N_ / 32; ++w) {
    int j = w * 32 + lane;
    float e = s1v + s2b[j];
    e = (e >= 0.f) ? e : ALPHA_ * e;
    bool pred = arow[j] > 0;
    unsigned long long bl = __ballot(pred);        // wave32: low 32 bits valid
    if (lane == 0) mb[w] = (unsigned)bl;
    float tv = pred ? e : NEG_;
    if (tv > m) { l = l * __expf(m - tv) + 1.0f; m = tv; }
    else        { l += __expf(tv - m); }
  }
  for (int off = 16; off >= 1; off >>= 1) {
    float m2 = __shfl_xor(m, off, 32);
    float l2 = __shfl_xor(l, off, 32);
    float M  = fmaxf(m, m2);
    l = l * __expf(m - M) + l2 * __expf(m2 - M);
    m = M;
  }
  if (lane == 0) { mrow[row] = m; lrow[row] = l; }
}

// ------- Kernel 4: h_prime = attn @ Wh (WMMA, attn recomputed from bitmask), ------
// ------- then LDS reduce across 4 waves + LayerNorm + exact GELU epilogue. --------
__global__ __launch_bounds__(128) void hprime_kernel(
    const unsigned* __restrict__ mbits, const float* __restrict__ Wh,
    const float* __restrict__ s1, const float* __restrict__ s2,
    const float* __restrict__ mrow, const float* __restrict__ lrow,
    const float* __restrict__ gamma, const float* __restrict__ beta,
    float* __restrict__ out) {
  __shared__ float lds[4][16][FOUT];
  int tid  = threadIdx.x;
  int lane = tid & 31;
  int wave = tid >> 5;
  int itile = blockIdx.x & (N_ / 16 - 1);
  int b     = blockIdx.x >> 7;
  int i0    = itile * 16;
  int r16   = lane & 15;
  int koff  = (lane >> 4) << 1;
  int rowid = b * N_ + i0 + r16;

  float s1v  = s1[rowid];
  float mi   = mrow[rowid];
  float invl = 1.0f / lrow[rowid];
  const unsigned* mb = mbits + (size_t)rowid * (N_ / 32);
  const float* s2b   = s2 + (size_t)b * N_;
  const float* Whb   = Wh + (size_t)b * N_ * FOUT;

  v8f acc[8];
#pragma unroll
  for (int ot = 0; ot < 8; ++ot) acc[ot] = (v8f){};

  int jbeg = wave * (N_ / 4);
  int jend = jbeg + (N_ / 4);
  for (int j0 = jbeg; j0 < jend; j0 += 4) {
    int j = j0 + koff;
    unsigned wbits = mb[j0 >> 5];        // bits for this 32-j window (L2-hit)
    v2f a;
#pragma unroll
    for (int v = 0; v < 2; ++v) {
      int jj = j + v;
      float e = s1v + s2b[jj];
      e = (e >= 0.f) ? e : ALPHA_ * e;
      float tv = ((wbits >> ((j0 & 31) + koff + v)) & 1u) ? e : NEG_;
      a[v] = __expf(tv - mi) * invl;
    }
    const float* w0 = Whb + (size_t)j * FOUT + r16;
    const float* w1 = w0 + FOUT;
#pragma unroll
    for (int ot = 0; ot < 8; ++ot) {
      v2f bb;
      bb.x = w0[ot * 16];
      bb.y = w1[ot * 16];
      acc[ot] = __builtin_amdgcn_wmma_f32_16x16x4_f32(false, a, false, bb,
                                                      (short)0, acc[ot],
                                                      false, false);
    }
  }

  // Spill each wave's 16x128 partial tile to LDS (C/D layout: lane%16 = N,
  // VGPR r -> M = r + 8*(lane>=16)).
#pragma unroll
  for (int ot = 0; ot < 8; ++ot)
#pragma unroll
    for (int r = 0; r < 8; ++r) {
      int M = r + ((lane >> 4) << 3);
      lds[wave][M][ot * 16 + r16] = acc[ot][r];
    }
  __syncthreads();

  // 128 threads: 8 threads per output row (16 rows), 16 cols each.
  int row16 = tid >> 3;
  int seg   = tid & 7;
  float vals[16];
  float s = 0.f, ss = 0.f;
#pragma unroll
  for (int c = 0; c < 16; ++c) {
    int col = seg * 16 + c;
    float x = lds[0][row16][col] + lds[1][row16][col] +
              lds[2][row16][col] + lds[3][row16][col];
    vals[c] = x;
    s += x;
    ss += x * x;
  }
  for (int off = 4; off >= 1; off >>= 1) {
    s  += __shfl_xor(s, off, 8);
    ss += __shfl_xor(ss, off, 8);
  }
  float mu   = s * (1.0f / FOUT);
  float var  = ss * (1.0f / FOUT) - mu * mu;
  float rstd = rsqrtf(var + EPS_);
  float* orow = out + (size_t)(b * N_ + i0 + row16) * FOUT;
#pragma unroll
  for (int c = 0; c < 16; ++c) {
    int col = seg * 16 + c;
    float x = (vals[c] - mu) * rstd * gamma[col] + beta[col];
    orow[col] = 0.5f * x * (1.0f + erff(x * 0.70710678118654752f));
  }
}

extern "C" void kernel_launch(void* const* d_in, const int* in_sizes, int n_in,
                              void* d_out, int out_size, void* d_ws, size_t ws_size,
                              hipStream_t stream) {
  const float* h     = (const float*)d_in[0];
  const int*   adj   = (const int*)d_in[1];
  const float* W     = (const float*)d_in[2];
  const float* a     = (const float*)d_in[3];
  const float* gamma = (const float*)d_in[4];
  const float* beta  = (const float*)d_in[5];
  float* out = (float*)d_out;

  float* ws   = (float*)d_ws;
  float* Wh   = ws;                                   // B*N*FOUT floats (4MB)
  float* s1   = Wh + (size_t)B_ * N_ * FOUT;          // B*N
  float* s2   = s1 + B_ * N_;
  float* mrow = s2 + B_ * N_;
  float* lrow = mrow + B_ * N_;
  unsigned* mbits = (unsigned*)(lrow + B_ * N_);      // B*N*(N/32) words (2MB)

  // K1: 4096 tiles, 4 waves/block
  wh_kernel<<<(B_ * (N_ / 16) * 8) / 4, 128, 0, stream>>>(h, W, Wh);
  // K2: wave per node
  s12_kernel<<<(B_ * N_ * 32) / 256, 256, 0, stream>>>(Wh, a, s1, s2);
  // K3: wave per row (single pass over adj, packs bitmask)
  stats_kernel<<<(B_ * N_) / 8, 256, 0, stream>>>(adj, s1, s2, mrow, lrow, mbits);
  // K4: block per (b, 16-row tile), adj never re-read
  hprime_kernel<<<B_ * (N_ / 16), 128, 0, stream>>>(mbits, Wh, s1, s2, mrow, lrow,
                                                    gamma, beta, out);
}